// CTransformer_73383811219777
// MI455X (gfx1250) — compile-verified
//
#include <hip/hip_runtime.h>
#include <hip/hip_bf16.h>
#include <math.h>

// ---------------------------------------------------------------------------
// CTransformer on MI455X (gfx1250): fp32 WMMA (16x16x4) for all GEMMs,
// VALU+LDS kernel for the CPB relative-position MLP, LDS reductions for
// layernorm/softmax. N=401 tokens padded to NP=416 (26 tiles of 16).
// Key fix vs prev round: all WMMA K-loops use a wave-uniform counter so the
// compiler keeps EXEC all-ones around v_wmma (ISA requirement).
// ---------------------------------------------------------------------------

typedef float v2f __attribute__((ext_vector_type(2)));
typedef float v8f __attribute__((ext_vector_type(8)));

#define D        512
#define NVALID   401      // 1 cls + 400 tokens (400 is a perfect square: no wrap pad)
#define NP       416      // padded to 26 * 16
#define NT       26       // row/col tiles of the token dimension
#define HEADS    8
#define HD       64
#define SCALE    0.125f   // 64^-0.5
#define LN_EPS   1e-5f

static __device__ __forceinline__ v8f wmma4(v2f a, v2f b, v8f c) {
  // D = A(16x4,f32) * B(4x16,f32) + C(16x16,f32)
  return __builtin_amdgcn_wmma_f32_16x16x4_f32(
      /*neg_a=*/false, a, /*neg_b=*/false, b,
      /*c_mod=*/(short)0, c, /*reuse_a=*/false, /*reuse_b=*/false);
}

// ---------------------------------------------------------------------------
// prep: X[0]=cls_token, X[401..415]=0, padded coords CP (row0=0, pad=0)
// ---------------------------------------------------------------------------
__global__ void prep_kernel(const float* __restrict__ cls,
                            const float* __restrict__ coords,
                            float* __restrict__ X, float* __restrict__ CP) {
  int gtid = blockIdx.x * blockDim.x + threadIdx.x;
  int gsz  = gridDim.x * blockDim.x;
  for (int c = gtid; c < D; c += gsz) X[c] = cls[c];
  for (int i = gtid; i < (NP - NVALID) * D; i += gsz) X[NVALID * D + i] = 0.f;
  for (int i = gtid; i < NP * 2; i += gsz) {
    float v = 0.f;
    if (i >= 2 && i < 2 + 400 * 2) v = coords[i - 2];
    CP[i] = v;
  }
}

// ---------------------------------------------------------------------------
// Generic fp32 WMMA GEMM: C = act(A(M,K) @ B(K,N) + bias) [+ C_old], with
// rows >= validRows forced to 0. One wave per 16x16 tile, 8 waves / block.
// Compile-time RELU/RESID keep the epilogue branch-free.
// ---------------------------------------------------------------------------
template <bool RELU, bool RESID>
__global__ __launch_bounds__(256) void gemm_wmma_f32(
    const float* __restrict__ A, int lda,
    const float* __restrict__ B, int ldb,
    const float* __restrict__ bias,
    float* __restrict__ C, int ldc,
    int K, int Mtiles, int Ntiles, int validRows) {
  int wave = threadIdx.x >> 5, lane = threadIdx.x & 31;
  int tile = blockIdx.x * 8 + wave;
  if (tile >= Mtiles * Ntiles) return;          // wave-uniform: EXEC stays full
  int ti = tile / Ntiles, tj = tile - ti * Ntiles;
  int half = lane >> 4, ln = lane & 15;
  int row = ti * 16 + ln;
  int col = tj * 16 + ln;
  const float* Ar = A + (size_t)row * lda + 2 * half;   // lane's K-pair base
  const float* Bc = B + (size_t)(2 * half) * ldb + col;
  __builtin_prefetch(Ar, 0, 3);
  __builtin_prefetch(Bc, 0, 3);
  v8f acc = {0.f, 0.f, 0.f, 0.f, 0.f, 0.f, 0.f, 0.f};
#pragma unroll 4
  for (int kc = 0; kc < K; kc += 4) {           // uniform trip count
    v2f a = *(const v2f*)(Ar + kc);             // A[row][kc+2h .. +1]  (b64)
    v2f b;
    b.x = Bc[(size_t)kc * ldb];                 // B[kc+2h][col]
    b.y = Bc[(size_t)(kc + 1) * ldb];
    acc = wmma4(a, b, acc);
  }
  float bv = bias[col];
#pragma unroll
  for (int r = 0; r < 8; ++r) {
    int grow = ti * 16 + r + 8 * half;
    float v = acc[r] + bv;
    if (RELU) v = fmaxf(v, 0.f);
    size_t idx = (size_t)grow * ldc + col;
    if (RESID) v += C[idx];
    if (grow >= validRows) v = 0.f;
    C[idx] = v;
  }
}

// ---------------------------------------------------------------------------
// layernorm per row (512 elems, 256 threads/row); pad rows -> 0
// ---------------------------------------------------------------------------
__global__ __launch_bounds__(256) void layernorm_rows(
    const float* __restrict__ X, float* __restrict__ XN,
    const float* __restrict__ g, const float* __restrict__ b) {
  __shared__ float red[256];
  int row = blockIdx.x, t = threadIdx.x;
  if (row >= NVALID) {
    XN[(size_t)row * D + t] = 0.f;
    XN[(size_t)row * D + t + 256] = 0.f;
    return;
  }
  const float* x = X + (size_t)row * D;
  float a0 = x[t], a1 = x[t + 256];
  red[t] = a0 + a1;
  __syncthreads();
  for (int s = 128; s > 0; s >>= 1) { if (t < s) red[t] += red[t + s]; __syncthreads(); }
  float m = red[0] * (1.f / D);
  __syncthreads();
  float d0 = a0 - m, d1 = a1 - m;
  red[t] = d0 * d0 + d1 * d1;
  __syncthreads();
  for (int s = 128; s > 0; s >>= 1) { if (t < s) red[t] += red[t + s]; __syncthreads(); }
  float var = red[0] * (1.f / D);
  float rs = rsqrtf(var + LN_EPS);
  XN[(size_t)row * D + t]       = d0 * rs * g[t] + b[t];
  XN[(size_t)row * D + t + 256] = d1 * rs * g[t + 256] + b[t + 256];
}

// ---------------------------------------------------------------------------
// CPB bias: for each pair (i,j): rel = normalize(c_i - c_j);
// BIAS[h][i][j] = relu(rel @ w1 + b1) @ w2[:,h] + b2[h].  Weights in LDS.
// One thread per pair; NP*NP = 173056 = 676 * 256 exactly.
// ---------------------------------------------------------------------------
__global__ __launch_bounds__(256) void cpb_kernel(
    const float* __restrict__ CP,
    const float* __restrict__ w1, const float* __restrict__ b1,
    const float* __restrict__ w2, const float* __restrict__ b2,
    float* __restrict__ BIAS) {
  __shared__ float sw1[2 * D];   // w1 row-major (2,512)
  __shared__ float sb1[D];
  __shared__ float sw2[D * HEADS];
  int t = threadIdx.x;
  for (int i = t; i < 2 * D; i += 256) sw1[i] = w1[i];
  for (int i = t; i < D; i += 256) sb1[i] = b1[i];
  for (int i = t; i < D * HEADS; i += 256) sw2[i] = w2[i];
  __syncthreads();

  int pair = blockIdx.x * 256 + t;
  int i = pair / NP, j = pair - i * NP;
  float rx = CP[2 * i] - CP[2 * j];
  float ry = CP[2 * i + 1] - CP[2 * j + 1];
  float inv = 1.f / (sqrtf(rx * rx + ry * ry) + 1e-6f);
  rx *= inv; ry *= inv;
  float acc[HEADS];
#pragma unroll
  for (int h = 0; h < HEADS; ++h) acc[h] = 0.f;
  for (int k = 0; k < D; ++k) {
    float hk = fmaxf(fmaf(rx, sw1[k], fmaf(ry, sw1[D + k], sb1[k])), 0.f);
#pragma unroll
    for (int h = 0; h < HEADS; ++h) acc[h] = fmaf(hk, sw2[k * HEADS + h], acc[h]);
  }
#pragma unroll
  for (int h = 0; h < HEADS; ++h)
    BIAS[((size_t)h * NP + i) * NP + j] = acc[h] + b2[h];
}

// ---------------------------------------------------------------------------
// Scores: S[h][i][j] = SCALE * (Q_h[i] . K_h[j]) + BIAS[h][i][j]; mask j>=401.
// QKV row layout (1536): [Q(8x64) | K(8x64) | V(8x64)].  8*26*26 = 5408 tiles.
// ---------------------------------------------------------------------------
__global__ __launch_bounds__(256) void attn_scores(
    const float* __restrict__ QKV, const float* __restrict__ BIAS,
    float* __restrict__ S) {
  int wave = threadIdx.x >> 5, lane = threadIdx.x & 31;
  int tile = blockIdx.x * 8 + wave;
  if (tile >= HEADS * NT * NT) return;
  int h = tile / (NT * NT);
  int rem = tile - h * (NT * NT);
  int ti = rem / NT, tj = rem - ti * NT;
  int half = lane >> 4, ln = lane & 15;
  int row = ti * 16 + ln, col = tj * 16 + ln;
  const float* Qr = QKV + (size_t)row * 1536 + h * HD + 2 * half;
  const float* Kc = QKV + (size_t)col * 1536 + D + h * HD + 2 * half;
  v8f acc = {0.f, 0.f, 0.f, 0.f, 0.f, 0.f, 0.f, 0.f};
#pragma unroll
  for (int kc = 0; kc < HD; kc += 4) {          // uniform trip count
    v2f a = *(const v2f*)(Qr + kc);
    v2f b = *(const v2f*)(Kc + kc);
    acc = wmma4(a, b, acc);
  }
#pragma unroll
  for (int r = 0; r < 8; ++r) {
    int grow = ti * 16 + r + 8 * half;
    size_t idx = ((size_t)h * NP + grow) * NP + col;
    float v = acc[r] * SCALE + BIAS[idx];
    if (col >= NVALID) v = -1e30f;
    S[idx] = v;
  }
}

// ---------------------------------------------------------------------------
// Row softmax over 416 cols (masked cols carry -1e30 -> prob exactly 0)
// ---------------------------------------------------------------------------
__global__ __launch_bounds__(256) void softmax_rows(float* __restrict__ S) {
  __shared__ float red[256];
  int t = threadIdx.x;
  float* p = S + (size_t)blockIdx.x * NP;   // blockIdx.x = h*NP + row
  float v0 = p[t];
  float v1 = (t + 256 < NP) ? p[t + 256] : -1e30f;
  red[t] = fmaxf(v0, v1);
  __syncthreads();
  for (int s = 128; s > 0; s >>= 1) { if (t < s) red[t] = fmaxf(red[t], red[t + s]); __syncthreads(); }
  float mx = red[0];
  __syncthreads();
  float e0 = __expf(v0 - mx);
  float e1 = (t + 256 < NP) ? __expf(v1 - mx) : 0.f;
  red[t] = e0 + e1;
  __syncthreads();
  for (int s = 128; s > 0; s >>= 1) { if (t < s) red[t] += red[t + s]; __syncthreads(); }
  float inv = 1.f / red[0];
  p[t] = e0 * inv;
  if (t + 256 < NP) p[t + 256] = e1 * inv;
}

// ---------------------------------------------------------------------------
// O[i][h*64+d] = sum_j P[h][i][j] * V_h[j][d].   8*26*4 = 832 tiles.
// ---------------------------------------------------------------------------
__global__ __launch_bounds__(256) void attn_out(
    const float* __restrict__ S, const float* __restrict__ QKV,
    float* __restrict__ O) {
  int wave = threadIdx.x >> 5, lane = threadIdx.x & 31;
  int tile = blockIdx.x * 8 + wave;
  if (tile >= HEADS * NT * 4) return;
  int h = tile / (NT * 4);
  int rem = tile - h * (NT * 4);
  int ti = rem / 4, td = rem - ti * 4;
  int half = lane >> 4, ln = lane & 15;
  int row = ti * 16 + ln, col = td * 16 + ln;        // col in [0,64)
  const float* Pr = S + ((size_t)h * NP + row) * NP + 2 * half;
  const float* Vc = QKV + (size_t)(2 * half) * 1536 + 2 * D + h * HD + col;
  __builtin_prefetch(Pr, 0, 3);
  v8f acc = {0.f, 0.f, 0.f, 0.f, 0.f, 0.f, 0.f, 0.f};
#pragma unroll 4
  for (int kc = 0; kc < NP; kc += 4) {          // uniform trip count
    v2f a = *(const v2f*)(Pr + kc);
    v2f b;
    b.x = Vc[(size_t)kc * 1536];
    b.y = Vc[(size_t)(kc + 1) * 1536];
    acc = wmma4(a, b, acc);
  }
#pragma unroll
  for (int r = 0; r < 8; ++r) {
    int grow = ti * 16 + r + 8 * half;
    O[(size_t)grow * D + h * HD + col] = acc[r];
  }
}

// ---------------------------------------------------------------------------
// Head: LN(X[0]) @ fc2_w + fc2_b  -> out[2]
// ---------------------------------------------------------------------------
__global__ __launch_bounds__(256) void head_kernel(
    const float* __restrict__ X,
    const float* __restrict__ g, const float* __restrict__ b,
    const float* __restrict__ w, const float* __restrict__ bias,
    float* __restrict__ out) {
  __shared__ float red[256];
  __shared__ float xn[D];
  int t = threadIdx.x;
  float a0 = X[t], a1 = X[t + 256];
  red[t] = a0 + a1;
  __syncthreads();
  for (int s = 128; s > 0; s >>= 1) { if (t < s) red[t] += red[t + s]; __syncthreads(); }
  float m = red[0] * (1.f / D);
  __syncthreads();
  float d0 = a0 - m, d1 = a1 - m;
  red[t] = d0 * d0 + d1 * d1;
  __syncthreads();
  for (int s = 128; s > 0; s >>= 1) { if (t < s) red[t] += red[t + s]; __syncthreads(); }
  float rs = rsqrtf(red[0] * (1.f / D) + LN_EPS);
  xn[t]       = d0 * rs * g[t] + b[t];
  xn[t + 256] = d1 * rs * g[t + 256] + b[t + 256];
  __syncthreads();
  for (int cls = 0; cls < 2; ++cls) {
    __syncthreads();
    red[t] = xn[t] * w[t * 2 + cls] + xn[t + 256] * w[(t + 256) * 2 + cls];
    __syncthreads();
    for (int s = 128; s > 0; s >>= 1) { if (t < s) red[t] += red[t + s]; __syncthreads(); }
    if (t == 0) out[cls] = red[0] + bias[cls];
  }
}

// ---------------------------------------------------------------------------
// Host launcher
// ---------------------------------------------------------------------------
extern "C" void kernel_launch(void* const* d_in, const int* in_sizes, int n_in,
                              void* d_out, int out_size, void* d_ws, size_t ws_size,
                              hipStream_t stream) {
  (void)in_sizes; (void)n_in; (void)out_size; (void)ws_size;
  const float* h_in   = (const float*)d_in[0];   // (400,1024)
  const float* coords = (const float*)d_in[1];   // (400,2)
  const float* fc1_w  = (const float*)d_in[2];   // (1024,512)
  const float* fc1_b  = (const float*)d_in[3];
  const float* cls    = (const float*)d_in[4];   // (512,)
  struct LayerW {
    const float *ng, *nb, *cw1, *cb1, *cw2, *cb2, *qw, *qb, *pw, *pb;
  } L[2];
  for (int l = 0; l < 2; ++l) {
    int base = 5 + 10 * l;
    L[l].ng  = (const float*)d_in[base + 0];
    L[l].nb  = (const float*)d_in[base + 1];
    L[l].cw1 = (const float*)d_in[base + 2];
    L[l].cb1 = (const float*)d_in[base + 3];
    L[l].cw2 = (const float*)d_in[base + 4];
    L[l].cb2 = (const float*)d_in[base + 5];
    L[l].qw  = (const float*)d_in[base + 6];
    L[l].qb  = (const float*)d_in[base + 7];
    L[l].pw  = (const float*)d_in[base + 8];
    L[l].pb  = (const float*)d_in[base + 9];
  }
  const float* norm_g = (const float*)d_in[25];
  const float* norm_b = (const float*)d_in[26];
  const float* fc2_w  = (const float*)d_in[27];
  const float* fc2_b  = (const float*)d_in[28];
  float* out = (float*)d_out;

  // Workspace layout (floats)
  float* ws   = (float*)d_ws;
  float* X    = ws;                                   // NP*512
  float* XN   = X    + (size_t)NP * D;                // NP*512
  float* QKV  = XN   + (size_t)NP * D;                // NP*1536
  float* O    = QKV  + (size_t)NP * 3 * D;            // NP*512
  float* CP   = O    + (size_t)NP * D;                // NP*2
  float* BIAS = CP   + (size_t)NP * 2;                // 8*NP*NP
  float* S    = BIAS + (size_t)HEADS * NP * NP;       // 8*NP*NP

  // 1) prep: cls row, zero pad rows, padded coords
  prep_kernel<<<40, 256, 0, stream>>>(cls, coords, X, CP);

  // 2) fc1 + ReLU: (400x1024)@(1024x512) -> X rows 1..400
  gemm_wmma_f32<true, false><<<(25 * 32) / 8, 256, 0, stream>>>(
      h_in, 1024, fc1_w, D, fc1_b, X + D, D, 1024, 25, 32, /*validRows=*/400);

  for (int l = 0; l < 2; ++l) {
    // a) layernorm -> XN (pad rows zeroed)
    layernorm_rows<<<NP, 256, 0, stream>>>(X, XN, L[l].ng, L[l].nb);
    // b) qkv GEMM: (416x512)@(512x1536)
    gemm_wmma_f32<false, false><<<(NT * 96) / 8, 256, 0, stream>>>(
        XN, D, L[l].qw, 3 * D, L[l].qb, QKV, 3 * D, D, NT, 96, NP);
    // c) CPB relative-position bias
    cpb_kernel<<<(NP * NP) / 256, 256, 0, stream>>>(
        CP, L[l].cw1, L[l].cb1, L[l].cw2, L[l].cb2, BIAS);
    // d) scores + bias + mask
    attn_scores<<<(HEADS * NT * NT) / 8, 256, 0, stream>>>(QKV, BIAS, S);
    // e) softmax
    softmax_rows<<<HEADS * NP, 256, 0, stream>>>(S);
    // f) P @ V
    attn_out<<<(HEADS * NT * 4) / 8, 256, 0, stream>>>(S, QKV, O);
    // g) proj + residual (pad rows forced to 0)
    gemm_wmma_f32<false, true><<<(NT * 32) / 8, 256, 0, stream>>>(
        O, D, L[l].pw, D, L[l].pb, X, D, D, NT, 32, /*validRows=*/NVALID);
  }

  // 3) final LN on CLS row + fc2
  head_kernel<<<1, 256, 0, stream>>>(X, norm_g, norm_b, fc2_w, fc2_b, out);
}